// PaiNN_43774306680866
// MI455X (gfx1250) — compile-verified
//
#include <hip/hip_runtime.h>
#include <hip/hip_bf16.h>
#include <math.h>

// ---------------------------------------------------------------------------
// PaiNN forward for MI455X (gfx1250), fp32 via V_WMMA_F32_16X16X4_F32.
// ---------------------------------------------------------------------------

#define CN 10000   // nodes
#define CH 128     // hidden
#define CE 160000  // edges
#define CR 64      // rbf
#define CL 6       // layers

#define SSILU_SCALE 1.6666666666666667f
#define INV_SQRT_3  0.5773502691896258f
#define INV_SQRT_H  0.08838834764831845f
#define INV_SQRT_2  0.7071067811865476f

typedef float v2f __attribute__((ext_vector_type(2)));
typedef float v8f __attribute__((ext_vector_type(8)));

__device__ __forceinline__ float ssilu_f(float x) {
    return (x / (1.0f + __expf(-x))) * SSILU_SCALE;
}

static inline int cdiv(int a, int b) { return (a + b - 1) / b; }

// ---------------------------------------------------------------------------
// Generic fp32 WMMA GEMM:  C[M x Nc] = act(A[M x K] @ B[K x Nc] + bias)
// Block = 256 threads (8 waves). Wave w computes rows [blk*128 + 16w, +16)
// and a 64-wide column strip (4 WMMA tiles) at blockIdx.y*64.
// B is staged through LDS in BK x 64 panels (float4 global loads, shared by
// all 8 waves); A fragments are aligned float2 global loads (each element
// touched exactly once per wave).  Requires K % 32 == 0 (here K in {64,128,256}).
// ACT: 0 = identity, 1 = scaled SiLU.
// ---------------------------------------------------------------------------
#define BK 32
template <int ACT>
__global__ __launch_bounds__(256) void gemm_wmma(
    const float* __restrict__ A, const float* __restrict__ B,
    const float* __restrict__ bias, float* __restrict__ C,
    int M, int K, int Nc)
{
    __shared__ float Bs[BK * 64];

    const int wave = threadIdx.x >> 5;
    const int lane = threadIdx.x & 31;
    const int lr   = lane & 15;   // row-in-tile (A) / col-in-tile (B,C)
    const int lh   = lane >> 4;   // half select (K pairs / row offset +8)
    const int row0 = blockIdx.x * 128 + wave * 16;
    const int col0 = blockIdx.y * 64;
    const int r    = row0 + lr;
    const bool rok = (r < M);

    v8f acc[4] = {};

    for (int k0 = 0; k0 < K; k0 += BK) {
        __syncthreads();
        // Cooperative stage of B[k0..k0+BK) x [col0..col0+64) into LDS.
        for (int i = threadIdx.x; i < (BK * 64) / 4; i += 256) {
            const int kk = (i * 4) >> 6;    // panel row
            const int cc = (i * 4) & 63;    // panel col (multiple of 4)
            const int gc = col0 + cc;
            float4 v;
            if (gc + 3 < Nc) {
                v = *(const float4*)&B[(size_t)(k0 + kk) * Nc + gc];
            } else {
                v.x = (gc + 0 < Nc) ? B[(size_t)(k0 + kk) * Nc + gc + 0] : 0.0f;
                v.y = (gc + 1 < Nc) ? B[(size_t)(k0 + kk) * Nc + gc + 1] : 0.0f;
                v.z = (gc + 2 < Nc) ? B[(size_t)(k0 + kk) * Nc + gc + 2] : 0.0f;
                v.w = (gc + 3 < Nc) ? B[(size_t)(k0 + kk) * Nc + gc + 3] : 0.0f;
            }
            *(float4*)&Bs[kk * 64 + cc] = v;
        }
        __syncthreads();

#pragma unroll
        for (int k = 0; k < BK; k += 4) {
            const int ka = k + lh * 2;      // this lane's K pair within panel
            v2f a;
            if (rok) {
                const float2 av = *(const float2*)&A[(size_t)r * K + k0 + ka];
                a.x = av.x; a.y = av.y;
            } else {
                a.x = 0.0f; a.y = 0.0f;
            }
#pragma unroll
            for (int t = 0; t < 4; ++t) {
                const int c = t * 16 + lr;
                v2f b;
                b.x = Bs[ka * 64 + c];
                b.y = Bs[(ka + 1) * 64 + c];
                acc[t] = __builtin_amdgcn_wmma_f32_16x16x4_f32(
                    false, a, false, b, (short)0, acc[t], false, false);
            }
        }
    }

#pragma unroll
    for (int t = 0; t < 4; ++t) {
        const int c = col0 + t * 16 + lr;
        if (c >= Nc) continue;
        const float bv = bias ? bias[c] : 0.0f;
#pragma unroll
        for (int j = 0; j < 8; ++j) {
            const int rr = row0 + j + 8 * lh;   // C layout: VGPR j -> M=j (+8 upper half)
            if (rr < M) {
                float v = acc[t][j] + bv;
                if (ACT == 1) v = ssilu_f(v);
                C[(size_t)rr * Nc + c] = v;
            }
        }
    }
}

// ---------------------------------------------------------------------------
// LayerNorm over H=128 per node. One block of 128 threads per node.
// ---------------------------------------------------------------------------
__global__ __launch_bounds__(128) void lnorm_kernel(
    const float* __restrict__ x, const float* __restrict__ w,
    const float* __restrict__ b, float* __restrict__ out)
{
    __shared__ float red[128];
    const int n = blockIdx.x;
    const int h = threadIdx.x;
    const float v = x[(size_t)n * CH + h];
    red[h] = v; __syncthreads();
    for (int s = 64; s > 0; s >>= 1) { if (h < s) red[h] += red[h + s]; __syncthreads(); }
    const float mean = red[0] * (1.0f / CH);
    __syncthreads();
    const float d = v - mean;
    red[h] = d * d; __syncthreads();
    for (int s = 64; s > 0; s >>= 1) { if (h < s) red[h] += red[h + s]; __syncthreads(); }
    const float var = red[0] * (1.0f / CH);
    out[(size_t)n * CH + h] = d * rsqrtf(var + 1e-5f) * w[h] + b[h];
}

// ---------------------------------------------------------------------------
// Fused edge kernel: for 16 edges per block
//   rbfh = edge_rbf @ rbf_w + rbf_b          (WMMA, 16x64 @ 64x384)
//   m    = xh[src] * rbfh                    (gather in WMMA epilogue -> LDS)
//   dx[dst]   += m[:, :H]                    (atomic scatter)
//   dvec[dst] += (vec[src]*m2 + m3*edge_vec) * INV_SQRT_H
// ---------------------------------------------------------------------------
#define ALDS 66  // padded stride for the 16x64 A tile (bank-conflict avoidance)
__global__ __launch_bounds__(256) void edge_kernel(
    const float* __restrict__ edge_rbf,  // E x 64
    const float* __restrict__ rbf_w,     // 64 x 384
    const float* __restrict__ rbf_b,     // 384
    const float* __restrict__ xh,        // N x 384
    const float* __restrict__ vecbuf,    // N x 3 x 128
    const float* __restrict__ edge_vec,  // E x 3
    const int*   __restrict__ src,
    const int*   __restrict__ dst,
    float* __restrict__ dx,              // N x 128 (atomic)
    float* __restrict__ dvec)            // N x 3 x 128 (atomic)
{
    __shared__ float Atile[16 * ALDS];
    __shared__ float mtile[16 * 384];
    __shared__ int   s_src[16], s_dst[16];
    __shared__ float s_ev[48];

    const int e0  = blockIdx.x * 16;
    const int tid = threadIdx.x;

    // Prefetch next block's streaming edge_rbf tile into cache hierarchy.
    if (e0 + 16 < CE) {
        __builtin_prefetch(edge_rbf + (size_t)(e0 + 16) * CR + tid * 4, 0, 1);
    }

    if (tid < 16) { s_src[tid] = src[e0 + tid]; s_dst[tid] = dst[e0 + tid]; }
    if (tid < 48) { s_ev[tid] = edge_vec[(size_t)e0 * 3 + tid]; }
    for (int i = tid; i < (16 * CR) / 4; i += 256) {
        const int e = (i * 4) >> 6, k = (i * 4) & 63;
        const float4 v = *(const float4*)&edge_rbf[(size_t)(e0 + e) * CR + k];
        Atile[e * ALDS + k + 0] = v.x;
        Atile[e * ALDS + k + 1] = v.y;
        Atile[e * ALDS + k + 2] = v.z;
        Atile[e * ALDS + k + 3] = v.w;
    }
    __syncthreads();

    const int wave = tid >> 5;
    const int lane = tid & 31;
    const int lr = lane & 15, lh = lane >> 4;

#pragma unroll
    for (int tt = 0; tt < 3; ++tt) {
        const int t   = wave * 3 + tt;   // 0..23 column tiles of 16
        const int col = t * 16 + lr;
        v8f acc = {};
        for (int k = 0; k < CR; k += 4) {
            const int ka = k + lh * 2;
            v2f a, b;
            a.x = Atile[lr * ALDS + ka];
            a.y = Atile[lr * ALDS + ka + 1];
            b.x = rbf_w[(size_t)ka       * 384 + col];
            b.y = rbf_w[(size_t)(ka + 1) * 384 + col];
            acc = __builtin_amdgcn_wmma_f32_16x16x4_f32(
                false, a, false, b, (short)0, acc, false, false);
        }
        const float bv = rbf_b[col];
#pragma unroll
        for (int j = 0; j < 8; ++j) {
            const int er = j + 8 * lh;
            const float rbfh = acc[j] + bv;
            const float m = xh[(size_t)s_src[er] * 384 + col] * rbfh;
            mtile[er * 384 + col] = m;
        }
    }
    __syncthreads();

    // Phase 2: split mx/m2/m3, build mvec, scatter-add.
    for (int i = tid; i < 16 * CH; i += 256) {
        const int er = i >> 7, h = i & 127;
        const int s = s_src[er], d = s_dst[er];
        const float mx = mtile[er * 384 + h];
        const float m2 = mtile[er * 384 + CH + h] * INV_SQRT_3;
        const float m3 = mtile[er * 384 + 2 * CH + h];
        atomicAdd(&dx[(size_t)d * CH + h], mx);
#pragma unroll
        for (int dd = 0; dd < 3; ++dd) {
            const float vv = vecbuf[((size_t)s * 3 + dd) * CH + h];
            const float mv = (vv * m2 + m3 * s_ev[er * 3 + dd]) * INV_SQRT_H;
            atomicAdd(&dvec[((size_t)d * 3 + dd) * CH + h], mv);
        }
    }
}

// x = (x + dx) * INV_SQRT_2 ; vec += dvec
__global__ __launch_bounds__(256) void accum_kernel(
    float* __restrict__ xbuf, const float* __restrict__ dx,
    float* __restrict__ vecbuf, const float* __restrict__ dvec)
{
    const int i = blockIdx.x * 256 + threadIdx.x;
    if (i >= CN * CH) return;
    xbuf[i] = (xbuf[i] + dx[i]) * INV_SQRT_2;
    const int n = i >> 7, h = i & 127;
#pragma unroll
    for (int dd = 0; dd < 3; ++dd) {
        const size_t idx = ((size_t)n * 3 + dd) * CH + h;
        vecbuf[idx] += dvec[idx];
    }
}

// vec_dot = sum_c v1*v2 * INV_SQRT_H ; vnorm = sqrt(sum_c v2^2 + 1e-8)
// also builds cat(x, vnorm) into xcat (N x 256).
__global__ __launch_bounds__(256) void vdotnorm_kernel(
    const float* __restrict__ vp,    // N x 3 x 256
    const float* __restrict__ xbuf,  // N x 128
    float* __restrict__ vdot,        // N x 128
    float* __restrict__ xcat)        // N x 256
{
    const int i = blockIdx.x * 256 + threadIdx.x;
    if (i >= CN * CH) return;
    const int n = i >> 7, h = i & 127;
    const float* vpn = vp + (size_t)n * 768;
    float d0 = 0.0f, nn = 0.0f;
#pragma unroll
    for (int dd = 0; dd < 3; ++dd) {
        const float a = vpn[dd * 256 + h];        // v1
        const float b = vpn[dd * 256 + CH + h];   // v2
        d0 += a * b; nn += b * b;
    }
    vdot[i] = d0 * INV_SQRT_H;
    xcat[(size_t)n * 256 + h]      = xbuf[i];
    xcat[(size_t)n * 256 + CH + h] = sqrtf(nn + 1e-8f);
}

// x += (xu1 + xu2*vec_dot)*INV_SQRT_2 ; vec += xu3 * v1
__global__ __launch_bounds__(256) void xvu_update_kernel(
    const float* __restrict__ h,    // N x 384
    const float* __restrict__ vdot, // N x 128
    const float* __restrict__ vp,   // N x 3 x 256
    float* __restrict__ xbuf, float* __restrict__ vecbuf)
{
    const int i = blockIdx.x * 256 + threadIdx.x;
    if (i >= CN * CH) return;
    const int n = i >> 7, hh = i & 127;
    const float xu1 = h[(size_t)n * 384 + hh];
    const float xu2 = h[(size_t)n * 384 + CH + hh];
    const float xu3 = h[(size_t)n * 384 + 2 * CH + hh];
    xbuf[i] += (xu1 + xu2 * vdot[i]) * INV_SQRT_2;
#pragma unroll
    for (int dd = 0; dd < 3; ++dd)
        vecbuf[((size_t)n * 3 + dd) * CH + hh] += xu3 * vp[(size_t)n * 768 + dd * 256 + hh];
}

// vec1 = ||w||_2 over channel axis ; cat = [xsrc, vec1]  (N x 2O)
__global__ __launch_bounds__(256) void norm3cat_kernel(
    const float* __restrict__ w,    // N x 3 x O
    const float* __restrict__ xsrc, // N x O
    float* __restrict__ cat,        // N x 2O
    int O)
{
    const int i = blockIdx.x * 256 + threadIdx.x;
    if (i >= CN * O) return;
    const int n = i / O, o = i - n * O;
    float s = 0.0f;
#pragma unroll
    for (int dd = 0; dd < 3; ++dd) {
        const float a = w[((size_t)n * 3 + dd) * O + o];
        s += a * a;
    }
    cat[(size_t)n * (2 * O) + o]     = xsrc[(size_t)n * O + o];
    cat[(size_t)n * (2 * O) + O + o] = sqrtf(s);
}

// fx = ssilu(h[:, :hout]) ; fv = h[:, hout:2*hout, broadcast] * vec2
__global__ __launch_bounds__(256) void gate_split_kernel(
    const float* __restrict__ h,     // N x Hh
    const float* __restrict__ vec2,  // N x 3 x hout
    float* __restrict__ fx,          // N x hout
    float* __restrict__ fv,          // N x 3 x hout
    int Hh, int hout)
{
    const int i = blockIdx.x * 256 + threadIdx.x;
    if (i >= CN * hout) return;
    const int n = i / hout, o = i - n * hout;
    fx[i] = ssilu_f(h[(size_t)n * Hh + o]);
    const float xv = h[(size_t)n * Hh + hout + o];
#pragma unroll
    for (int dd = 0; dd < 3; ++dd)
        fv[((size_t)n * 3 + dd) * hout + o] = xv * vec2[((size_t)n * 3 + dd) * hout + o];
}

// out[n] = [energy, forces(3)]
__global__ __launch_bounds__(256) void assemble_kernel(
    const float* __restrict__ energy, // N x 1
    const float* __restrict__ h2,     // N x 2  (xs, xv)
    const float* __restrict__ vec2b,  // N x 3 x 1
    float* __restrict__ out)          // N x 4
{
    const int n = blockIdx.x * 256 + threadIdx.x;
    if (n >= CN) return;
    out[(size_t)n * 4 + 0] = energy[n];
    const float xv = h2[(size_t)n * 2 + 1];
#pragma unroll
    for (int dd = 0; dd < 3; ++dd)
        out[(size_t)n * 4 + 1 + dd] = xv * vec2b[(size_t)n * 3 + dd];
}

// ---------------------------------------------------------------------------
extern "C" void kernel_launch(void* const* d_in, const int* in_sizes, int n_in,
                              void* d_out, int out_size, void* d_ws, size_t ws_size,
                              hipStream_t stream)
{
    const int N = CN, H = CH, E = CE, R = CR;

    const float* x_in     = (const float*)d_in[0];
    const float* vec_in   = (const float*)d_in[1];
    const int*   eidx     = (const int*)  d_in[2];
    const float* edge_rbf = (const float*)d_in[3];
    const float* edge_vec = (const float*)d_in[4];
    const float* ln_w   = (const float*)d_in[5];
    const float* ln_b   = (const float*)d_in[6];
    const float* xp1_w  = (const float*)d_in[7];
    const float* xp1_b  = (const float*)d_in[8];
    const float* xp2_w  = (const float*)d_in[9];
    const float* xp2_b  = (const float*)d_in[10];
    const float* rbf_w  = (const float*)d_in[11];
    const float* rbf_b  = (const float*)d_in[12];
    const float* vp_w   = (const float*)d_in[13];
    const float* xv1_w  = (const float*)d_in[14];
    const float* xv1_b  = (const float*)d_in[15];
    const float* xv2_w  = (const float*)d_in[16];
    const float* xv2_b  = (const float*)d_in[17];
    const float* we1    = (const float*)d_in[18];
    const float* be1    = (const float*)d_in[19];
    const float* we2    = (const float*)d_in[20];
    const float* be2    = (const float*)d_in[21];
    const float* g1_v1  = (const float*)d_in[22];
    const float* g1_v2  = (const float*)d_in[23];
    const float* g1_u1  = (const float*)d_in[24];
    const float* g1_u1b = (const float*)d_in[25];
    const float* g1_u2  = (const float*)d_in[26];
    const float* g1_u2b = (const float*)d_in[27];
    const float* g2_v1  = (const float*)d_in[28];
    const float* g2_v2  = (const float*)d_in[29];
    const float* g2_u1  = (const float*)d_in[30];
    const float* g2_u1b = (const float*)d_in[31];
    const float* g2_u2  = (const float*)d_in[32];
    const float* g2_u2b = (const float*)d_in[33];

    const int* src = eidx;
    const int* dst = eidx + E;

    // ---- workspace carve-up (floats) ----
    float* ws = (float*)d_ws;
    size_t off = 0;
    auto take = [&](size_t nf) { float* p = ws + off; off += nf; return p; };
    float* xbuf   = take((size_t)N * H);
    float* vecbuf = take((size_t)N * 3 * H);
    float* tbuf   = take((size_t)N * H);
    float* ubuf   = take((size_t)N * H);
    float* xh     = take((size_t)N * 3 * H);
    float* dx     = take((size_t)N * H);
    float* dvec   = take((size_t)N * 3 * H);
    float* vp     = take((size_t)N * 6 * H);
    float* vdot   = take((size_t)N * H);
    float* xcat   = take((size_t)N * 2 * H);
    float* hbuf   = take((size_t)N * 3 * H);
    // head-phase aliases (buffers dead after the layer loop):
    float* energy = dx;
    float* w1buf  = xh;
    float* vec2b1 = dvec;     // N x 3 x 64
    float* fxbuf  = ubuf;     // N x 64
    float* fvbuf  = vp;       // N x 3 x 64
    float* w2buf  = xh;       // N x 3 x 64 (w1 dead after norm3cat)
    float* vec2b2 = vdot;     // N x 3
    float* h2buf  = hbuf;     // N x 2 (h1 dead after gate_split)

    hipMemcpyAsync(xbuf, x_in, (size_t)N * H * sizeof(float),
                   hipMemcpyDeviceToDevice, stream);
    hipMemcpyAsync(vecbuf, vec_in, (size_t)N * 3 * H * sizeof(float),
                   hipMemcpyDeviceToDevice, stream);

    const int EW = N * H;                 // elementwise size
    const int EWB = cdiv(EW, 256);

    for (int l = 0; l < CL; ++l) {
        // node MLP: xh = ssilu(lnorm(x) @ xp1 + b) @ xp2 + b
        lnorm_kernel<<<N, 128, 0, stream>>>(xbuf, ln_w + l * H, ln_b + l * H, tbuf);
        gemm_wmma<1><<<dim3(cdiv(N, 128), cdiv(H, 64)), 256, 0, stream>>>(
            tbuf, xp1_w + (size_t)l * H * H, xp1_b + l * H, ubuf, N, H, H);
        gemm_wmma<0><<<dim3(cdiv(N, 128), cdiv(3 * H, 64)), 256, 0, stream>>>(
            ubuf, xp2_w + (size_t)l * H * 3 * H, xp2_b + (size_t)l * 3 * H, xh, N, H, 3 * H);

        // edge message + scatter
        hipMemsetAsync(dx, 0, (size_t)N * H * sizeof(float), stream);
        hipMemsetAsync(dvec, 0, (size_t)N * 3 * H * sizeof(float), stream);
        edge_kernel<<<E / 16, 256, 0, stream>>>(
            edge_rbf, rbf_w + (size_t)l * R * 3 * H, rbf_b + (size_t)l * 3 * H,
            xh, vecbuf, edge_vec, src, dst, dx, dvec);
        accum_kernel<<<EWB, 256, 0, stream>>>(xbuf, dx, vecbuf, dvec);

        // vector channel mixing: vp = vec @ vp_w   ((3N x H) @ (H x 2H))
        gemm_wmma<0><<<dim3(cdiv(3 * N, 128), cdiv(2 * H, 64)), 256, 0, stream>>>(
            vecbuf, vp_w + (size_t)l * H * 2 * H, nullptr, vp, 3 * N, H, 2 * H);
        vdotnorm_kernel<<<EWB, 256, 0, stream>>>(vp, xbuf, vdot, xcat);

        // h = ssilu(cat(x, vnorm) @ xv1 + b) @ xv2 + b
        gemm_wmma<1><<<dim3(cdiv(N, 128), cdiv(H, 64)), 256, 0, stream>>>(
            xcat, xv1_w + (size_t)l * 2 * H * H, xv1_b + l * H, tbuf, N, 2 * H, H);
        gemm_wmma<0><<<dim3(cdiv(N, 128), cdiv(3 * H, 64)), 256, 0, stream>>>(
            tbuf, xv2_w + (size_t)l * H * 3 * H, xv2_b + (size_t)l * 3 * H, hbuf, N, H, 3 * H);
        xvu_update_kernel<<<EWB, 256, 0, stream>>>(hbuf, vdot, vp, xbuf, vecbuf);
    }

    // ---- energy head ----
    gemm_wmma<1><<<dim3(cdiv(N, 128), 1), 256, 0, stream>>>(xbuf, we1, be1, tbuf, N, H, H / 2);
    gemm_wmma<0><<<dim3(cdiv(N, 128), 1), 256, 0, stream>>>(tbuf, we2, be2, energy, N, H / 2, 1);

    // ---- gated block 1 (hout = 64) ----
    gemm_wmma<0><<<dim3(cdiv(3 * N, 128), cdiv(H, 64)), 256, 0, stream>>>(
        vecbuf, g1_v1, nullptr, w1buf, 3 * N, H, H);
    gemm_wmma<0><<<dim3(cdiv(3 * N, 128), 1), 256, 0, stream>>>(
        vecbuf, g1_v2, nullptr, vec2b1, 3 * N, H, H / 2);
    norm3cat_kernel<<<cdiv(N * H, 256), 256, 0, stream>>>(w1buf, xbuf, xcat, H);
    gemm_wmma<1><<<dim3(cdiv(N, 128), cdiv(H, 64)), 256, 0, stream>>>(
        xcat, g1_u1, g1_u1b, tbuf, N, 2 * H, H);
    gemm_wmma<0><<<dim3(cdiv(N, 128), cdiv(H, 64)), 256, 0, stream>>>(
        tbuf, g1_u2, g1_u2b, hbuf, N, H, H);
    gate_split_kernel<<<cdiv(N * (H / 2), 256), 256, 0, stream>>>(
        hbuf, vec2b1, fxbuf, fvbuf, H, H / 2);

    // ---- gated block 2 (hout = 1) ----
    gemm_wmma<0><<<dim3(cdiv(3 * N, 128), 1), 256, 0, stream>>>(
        fvbuf, g2_v1, nullptr, w2buf, 3 * N, H / 2, H / 2);
    gemm_wmma<0><<<dim3(cdiv(3 * N, 128), 1), 256, 0, stream>>>(
        fvbuf, g2_v2, nullptr, vec2b2, 3 * N, H / 2, 1);
    norm3cat_kernel<<<cdiv(N * (H / 2), 256), 256, 0, stream>>>(w2buf, fxbuf, xcat, H / 2);
    gemm_wmma<1><<<dim3(cdiv(N, 128), 1), 256, 0, stream>>>(
        xcat, g2_u1, g2_u1b, tbuf, N, H, H / 2);
    gemm_wmma<0><<<dim3(cdiv(N, 128), 1), 256, 0, stream>>>(
        tbuf, g2_u2, g2_u2b, h2buf, N, H / 2, 2);

    assemble_kernel<<<cdiv(N, 256), 256, 0, stream>>>(energy, h2buf, vec2b2, (float*)d_out);
}